// CCEmbedder_52192442581720
// MI455X (gfx1250) — compile-verified
//
#include <hip/hip_runtime.h>

// ---------------------------------------------------------------------------
// CCEmbedder on MI455X (gfx1250): two HMC message-passing levels.
// Heavy work = dense neighborhood GEMMs A[M,K] @ B[K,F] with fp32 data
// (A matrices total ~640MB; ~1.2GB streamed overall -> memory bound at
// 8-32 FLOP/byte). Math on the fp32 WMMA pipe (V_WMMA_F32_16X16X4_F32);
// slab staging via GLOBAL_LOAD_ASYNC_TO_LDS_B128 with double buffering and
// s_wait_asynccnt when available; split-K with f32 global atomics; ReLU
// fused into finalize passes.
// ---------------------------------------------------------------------------

typedef float v2f __attribute__((ext_vector_type(2)));
typedef float v8f __attribute__((ext_vector_type(8)));

// D = A(16x4) * B(4x16) + C, fp32:
// (neg_a, A, neg_b, B, c_mod, C, reuse_a, reuse_b)
#define WMMA_F32X4(a, b, c) \
  __builtin_amdgcn_wmma_f32_16x16x4_f32(false, (a), false, (b), (short)0, (c), false, false)

#if defined(__has_builtin)
#if __has_builtin(__builtin_amdgcn_global_load_async_to_lds_b128) && \
    __has_builtin(__builtin_amdgcn_s_wait_asynccnt)
#define USE_ASYNC_LDS 1
#endif
#endif
#ifndef USE_ASYNC_LDS
#define USE_ASYNC_LDS 0
#endif

#if USE_ASYNC_LDS
// Param 1 is a generic, non-const pointer to a vector_size(16) int (per
// hipcc's diagnostic). Param 2: AS(3)-qualified pointer works whether the
// builtin wants AS3 (exact) or generic (legal AS3->generic widening).
typedef int v4i_vs __attribute__((vector_size(4 * sizeof(int))));
typedef __attribute__((address_space(3))) v4i_vs* lds_v4i_ptr;
#define ASYNC_LD_B128(gsrc, ldst) \
  __builtin_amdgcn_global_load_async_to_lds_b128((v4i_vs*)(gsrc), \
                                                 (lds_v4i_ptr)(ldst), 0, 0)
#endif

__device__ __forceinline__ v8f v8f_zero() {
  v8f z = {0.f, 0.f, 0.f, 0.f, 0.f, 0.f, 0.f, 0.f};
  return z;
}

// ---------------------------------------------------------------------------
// agg_kernel<F, TRANSA>: Cacc[M,F] += (TRANSA ? A^T : A) @ B[K,F]
//   !TRANSA: A stored row-major [M,K]  -> slab staged as sA[m][k], pitch 36
//    TRANSA: A stored row-major [K,M]  -> slab staged as sA[k][m], pitch 20
//   B slab sB[k][n], pitch 20 (F=16) / 80 (F=64): pitches chosen so the two
//   half-wave fragment reads hit disjoint LDS bank sets and every staged
//   transfer is a contiguous, 16B-aligned float4 (async b128-friendly).
// One wave per 16-row output tile per K-chunk; 8 WMMA f32 16x16x4 per slab,
// F=64 reuses each A slab for 4 N-tiles (32 FLOP per A-byte).
// ---------------------------------------------------------------------------
template <int F, bool TRANSA>
__launch_bounds__(32)
__global__ void agg_kernel(const float* __restrict__ A,
                           const float* __restrict__ B,
                           float* __restrict__ Cacc,
                           int M, int K, int kChunk) {
  constexpr int NT     = F / 16;
  constexpr int BK     = 32;
  constexpr int PA     = TRANSA ? 20 : 36;   // A slab row pitch (floats)
  constexpr int ROWS_A = TRANSA ? BK : 16;
  constexpr int PB     = (F == 16) ? 20 : 80;
  constexpr int SLAB_OPS = 4 + F / 4;        // async instrs per slab (A + B)

  __shared__ float sA[2][ROWS_A * PA];
  __shared__ float sB[2][BK * PB];

  const int lane = threadIdx.x;
  const int hi   = lane >> 4;    // 0: lanes 0-15, 1: lanes 16-31
  const int lm   = lane & 15;
  const int mt   = blockIdx.x * 16;
  const int kbeg = blockIdx.y * kChunk;
  int kend = kbeg + kChunk;
  if (kend > K) kend = K;        // K is always a multiple of 32

  v8f acc[NT];
#pragma unroll
  for (int t = 0; t < NT; ++t) acc[t] = v8f_zero();

  // Stage one BK-slab of A and B into LDS buffer `buf` (contiguous float4s).
  auto stage = [&](int buf, int kb) {
    // A slab: 128 float4 (16x32 or 32x16 floats)
#pragma unroll
    for (int e = lane; e < 128; e += 32) {
      int r, c4;
      size_t g;
      if (!TRANSA) {
        r  = e >> 3;             // 8 float4 per 32-float row
        c4 = (e & 7) << 2;
        g  = (size_t)(mt + r) * K + kb + c4;
      } else {
        r  = e >> 2;             // 4 float4 per 16-float row
        c4 = (e & 3) << 2;
        g  = (size_t)(kb + r) * M + mt + c4;
      }
      float* dst = &sA[buf][r * PA + c4];
#if USE_ASYNC_LDS
      ASYNC_LD_B128(A + g, dst);
#else
      *reinterpret_cast<float4*>(dst) =
          *reinterpret_cast<const float4*>(A + g);
#endif
    }
    // B slab: 32 x F floats = 8F float4
#pragma unroll
    for (int e = lane; e < 8 * F; e += 32) {
      const int k  = e / (F / 4);
      const int n4 = (e % (F / 4)) << 2;
      const size_t g = (size_t)(kb + k) * F + n4;
      float* dst = &sB[buf][k * PB + n4];
#if USE_ASYNC_LDS
      ASYNC_LD_B128(B + g, dst);
#else
      *reinterpret_cast<float4*>(dst) =
          *reinterpret_cast<const float4*>(B + g);
#endif
    }
  };

  // 8 WMMA k-steps over the slab in buffer `buf`.
  auto compute = [&](int buf) {
#pragma unroll
    for (int k0 = 0; k0 < BK; k0 += 4) {
      // A frag 16x4: lane holds M=lm; VGPR0 = K(k0+2*hi), VGPR1 = K(+1)
      v2f a;
      if (!TRANSA) {
        const float* pa = &sA[buf][lm * PA + k0 + 2 * hi];  // contiguous b64
        a.x = pa[0];
        a.y = pa[1];
      } else {
        a.x = sA[buf][(k0 + 2 * hi + 0) * PA + lm];
        a.y = sA[buf][(k0 + 2 * hi + 1) * PA + lm];
      }
#pragma unroll
      for (int t = 0; t < NT; ++t) {
        v2f b;  // B frag 4x16, mirrored layout: lane holds N=lm
        b.x = sB[buf][(k0 + 2 * hi + 0) * PB + t * 16 + lm];
        b.y = sB[buf][(k0 + 2 * hi + 1) * PB + t * 16 + lm];
        acc[t] = WMMA_F32X4(a, b, acc[t]);
      }
    }
  };

#if USE_ASYNC_LDS
  // Double-buffered async pipeline (single wave per workgroup -> counter
  // waits are sufficient; async loads complete in order).
  stage(0, kbeg);
  int ibuf = 0;
  for (int kb = kbeg; kb < kend; kb += BK) {
    if (kb + BK < kend) {
      stage(ibuf ^ 1, kb + BK);
      __builtin_amdgcn_s_wait_asynccnt(SLAB_OPS);  // slab `ibuf` has landed
    } else {
      __builtin_amdgcn_s_wait_asynccnt(0);
    }
    compute(ibuf);
    ibuf ^= 1;
  }
#else
  for (int kb = kbeg; kb < kend; kb += BK) {
    stage(0, kb);
    if (kb + BK < kend) {  // prefetch next A slab
      if (!TRANSA)
        __builtin_prefetch(A + (size_t)(mt + lm) * K + kb + BK + hi * 16, 0, 0);
      else
        __builtin_prefetch(A + (size_t)(kb + BK + lane) * M + mt, 0, 0);
    }
    __syncthreads();
    compute(0);
    __syncthreads();
  }
#endif

  // ---- split-K accumulate: C/D layout row = r + 8*hi, col = lm ----
#pragma unroll
  for (int t = 0; t < NT; ++t) {
#pragma unroll
    for (int r = 0; r < 8; ++r) {
      const int row = mt + r + 8 * hi;
      __hip_atomic_fetch_add(Cacc + (size_t)row * F + t * 16 + lm, acc[t][r],
                             __ATOMIC_RELAXED, __HIP_MEMORY_SCOPE_AGENT);
    }
  }
}

// ---------------------------------------------------------------------------
// proj_kernel<KD, ND>: C[M,ND] = X[M,KD] @ W[KD,ND]  (KD*ND == 1024)
// One wave per 16-row tile; W cached in LDS; A-frags read straight from
// global (compiler clauses them as b64 loads feeding WMMA).
// ---------------------------------------------------------------------------
template <int KD, int ND>
__launch_bounds__(32)
__global__ void proj_kernel(const float* __restrict__ X,
                            const float* __restrict__ W,
                            float* __restrict__ C) {
  constexpr int NT = ND / 16;
  __shared__ float sW[KD][ND + 1];

  const int lane = threadIdx.x;
  const int hi   = lane >> 4;
  const int lm   = lane & 15;
  const int mt   = blockIdx.x * 16;

#pragma unroll
  for (int e = lane; e < (KD * ND) / 4; e += 32) {
    const int k  = e / (ND / 4);
    const int n4 = (e % (ND / 4)) << 2;
    const float4 v = *reinterpret_cast<const float4*>(W + (size_t)k * ND + n4);
    sW[k][n4 + 0] = v.x; sW[k][n4 + 1] = v.y;
    sW[k][n4 + 2] = v.z; sW[k][n4 + 3] = v.w;
  }
  __syncthreads();

  v8f acc[NT];
#pragma unroll
  for (int t = 0; t < NT; ++t) acc[t] = v8f_zero();

#pragma unroll
  for (int k0 = 0; k0 < KD; k0 += 4) {
    v2f a;
    a.x = X[(size_t)(mt + lm) * KD + k0 + 2 * hi + 0];
    a.y = X[(size_t)(mt + lm) * KD + k0 + 2 * hi + 1];
#pragma unroll
    for (int t = 0; t < NT; ++t) {
      v2f b;
      b.x = sW[k0 + 2 * hi + 0][t * 16 + lm];
      b.y = sW[k0 + 2 * hi + 1][t * 16 + lm];
      acc[t] = WMMA_F32X4(a, b, acc[t]);
    }
  }

#pragma unroll
  for (int t = 0; t < NT; ++t)
#pragma unroll
    for (int r = 0; r < 8; ++r)
      C[(size_t)(mt + r + 8 * hi) * ND + t * 16 + lm] = acc[t][r];
}

// ---------------------------------------------------------------------------
// Elementwise helpers
// ---------------------------------------------------------------------------
__global__ void zero_kernel(float* __restrict__ p, int n) {
  int i = blockIdx.x * blockDim.x + threadIdx.x;
  const int stride = gridDim.x * blockDim.x;
  for (; i < n; i += stride) p[i] = 0.f;
}

__global__ void relu_inplace_kernel(float* __restrict__ p, int n) {
  int i = blockIdx.x * blockDim.x + threadIdx.x;
  const int stride = gridDim.x * blockDim.x;
  for (; i < n; i += stride) {
    const float v = p[i];
    p[i] = v > 0.f ? v : 0.f;
  }
}

__global__ void relu_copy_kernel(const float* __restrict__ s,
                                 float* __restrict__ d, int n) {
  int i = blockIdx.x * blockDim.x + threadIdx.x;
  const int stride = gridDim.x * blockDim.x;
  for (; i < n; i += stride) {
    const float v = s[i];
    d[i] = v > 0.f ? v : 0.f;
  }
}

// ---------------------------------------------------------------------------
// Host orchestration
// ---------------------------------------------------------------------------
template <int F, bool TR>
static void launch_agg(const float* A, const float* B, float* C, int M, int K,
                       hipStream_t s) {
  const int tiles = M / 16;
  int target = 8192 / tiles;              // aim for ~8k waves total
  if (target < 1) target = 1;
  if (target > 16) target = 16;
  const int kc  = (((K + target - 1) / target) + 31) & ~31;  // multiple of 32
  const int nch = (K + kc - 1) / kc;
  agg_kernel<F, TR><<<dim3(tiles, nch), 32, 0, s>>>(A, B, C, M, K, kc);
}

extern "C" void kernel_launch(void* const* d_in, const int* in_sizes, int n_in,
                              void* d_out, int out_size, void* d_ws,
                              size_t ws_size, hipStream_t stream) {
  (void)in_sizes; (void)n_in; (void)out_size; (void)ws_size;

  constexpr int N0 = 4000, N1 = 8000, N2 = 4000, F0 = 64, H = 16;

  const float* x0  = (const float*)d_in[0];
  const float* x1  = (const float*)d_in[1];
  const float* x2  = (const float*)d_in[2];
  const float* A00 = (const float*)d_in[3];   // [N0,N0]
  const float* A11 = (const float*)d_in[4];   // [N1,N1]
  const float* A22 = (const float*)d_in[5];   // [N2,N2]
  const float* A01 = (const float*)d_in[6];   // [N1,N0]
  const float* A12 = (const float*)d_in[7];   // [N2,N1]
  const float* w1_00 = (const float*)d_in[8];
  const float* w1_01 = (const float*)d_in[9];
  const float* w1_10 = (const float*)d_in[10];
  const float* w1_11 = (const float*)d_in[11];
  const float* w1_12 = (const float*)d_in[12];
  const float* w1_21 = (const float*)d_in[13];
  const float* w1_22 = (const float*)d_in[14];
  const float* w2_00 = (const float*)d_in[15];
  const float* w2_01 = (const float*)d_in[16];
  const float* w2_10 = (const float*)d_in[17];
  const float* w2_11 = (const float*)d_in[18];
  // w2_12 (19), w2_22 (21) unused: y2 is never returned
  const float* w2_21 = (const float*)d_in[20];

  float* ws = (float*)d_ws;
  // accumulators first (zeroed once, contiguous); y0a‖y1a matches d_out order
  float* h0  = ws;                       // N0*H   = 64000
  float* h1  = h0 + N0 * H;              // N1*H   = 128000
  float* h2  = h1 + N1 * H;              // N2*H   = 64000
  float* y0a = h2 + N2 * H;              // N0*F0  = 256000
  float* y1a = y0a + N0 * F0;            // N1*F0  = 512000
  const int accLen = (N0 + N1 + N2) * H + (N0 + N1) * F0;  // 1,024,000
  // projection buffers (fully overwritten each call)
  float* p00 = y1a + N1 * F0;
  float* p01 = p00 + N0 * H;
  float* p10 = p01 + N0 * H;
  float* p11 = p10 + N1 * H;
  float* p12 = p11 + N1 * H;
  float* p21 = p12 + N1 * H;
  float* p22 = p21 + N2 * H;
  float* q00 = p22 + N2 * H;
  float* q01 = q00 + N0 * F0;
  float* q10 = q01 + N0 * F0;
  float* q11 = q10 + N1 * F0;
  float* q21 = q11 + N1 * F0;

  // 0) zero the h/y accumulators
  zero_kernel<<<1024, 256, 0, stream>>>(ws, accLen);

  // 1) level-1 projections  p = x @ w1  ([n,64] @ [64,16])
  proj_kernel<64, 16><<<N0 / 16, 32, 0, stream>>>(x0, w1_00, p00);
  proj_kernel<64, 16><<<N0 / 16, 32, 0, stream>>>(x0, w1_01, p01);
  proj_kernel<64, 16><<<N1 / 16, 32, 0, stream>>>(x1, w1_10, p10);
  proj_kernel<64, 16><<<N1 / 16, 32, 0, stream>>>(x1, w1_11, p11);
  proj_kernel<64, 16><<<N1 / 16, 32, 0, stream>>>(x1, w1_12, p12);
  proj_kernel<64, 16><<<N2 / 16, 32, 0, stream>>>(x2, w1_21, p21);
  proj_kernel<64, 16><<<N2 / 16, 32, 0, stream>>>(x2, w1_22, p22);

  // 2) level-1 neighborhood aggregation (F = 16)
  launch_agg<16, false>(A00, p00, h0, N0, N0, stream);  // A00   @ p00
  launch_agg<16, true >(A01, p10, h0, N0, N1, stream);  // A01^T @ p10
  launch_agg<16, false>(A11, p11, h1, N1, N1, stream);  // A11   @ p11
  launch_agg<16, false>(A01, p01, h1, N1, N0, stream);  // A01   @ p01
  launch_agg<16, true >(A12, p21, h1, N1, N2, stream);  // A12^T @ p21
  launch_agg<16, false>(A22, p22, h2, N2, N2, stream);  // A22   @ p22
  launch_agg<16, false>(A12, p12, h2, N2, N1, stream);  // A12   @ p12

  // 3) ReLU on h0,h1,h2 (contiguous)
  relu_inplace_kernel<<<512, 256, 0, stream>>>(h0, (N0 + N1 + N2) * H);

  // 4) level-2 projections  q = h @ w2  ([n,16] @ [16,64]); skip y2 inputs
  proj_kernel<16, 64><<<N0 / 16, 32, 0, stream>>>(h0, w2_00, q00);
  proj_kernel<16, 64><<<N0 / 16, 32, 0, stream>>>(h0, w2_01, q01);
  proj_kernel<16, 64><<<N1 / 16, 32, 0, stream>>>(h1, w2_10, q10);
  proj_kernel<16, 64><<<N1 / 16, 32, 0, stream>>>(h1, w2_11, q11);
  proj_kernel<16, 64><<<N2 / 16, 32, 0, stream>>>(h2, w2_21, q21);

  // 5) level-2 neighborhood aggregation (F = 64), only y0 / y1
  launch_agg<64, false>(A00, q00, y0a, N0, N0, stream);  // A00   @ q00
  launch_agg<64, true >(A01, q10, y0a, N0, N1, stream);  // A01^T @ q10
  launch_agg<64, false>(A11, q11, y1a, N1, N1, stream);  // A11   @ q11
  launch_agg<64, false>(A01, q01, y1a, N1, N0, stream);  // A01   @ q01
  launch_agg<64, true >(A12, q21, y1a, N1, N2, stream);  // A12^T @ q21

  // 6) ReLU -> d_out (y0a‖y1a is contiguous and matches output order)
  relu_copy_kernel<<<1024, 256, 0, stream>>>(y0a, (float*)d_out,
                                             (N0 + N1) * F0);
}